// EnsembleRBF_87557203296990
// MI455X (gfx1250) — compile-verified
//
#include <hip/hip_runtime.h>

typedef __attribute__((ext_vector_type(2))) float v2f;
typedef __attribute__((ext_vector_type(8))) float v8f;

#define SIGMA2 0.0625f   // (0.25)^2

__global__ __launch_bounds__(256) void EnsembleRBF_wmma_kernel(
    const float* __restrict__ x,        // [N,2]
    const float* __restrict__ centers,  // [256,2]
    const float* __restrict__ weights,  // [5,256,2]
    float* __restrict__ out,            // [5,N,2]
    int N)
{
    // LDS staging: centers (2KB) + transposed/padded weights (16KB)
    __shared__ __align__(16) float lds_c[512];   // [c][xy]
    __shared__ __align__(16) float lds_w[4096];  // [c/2][j=0..15][parity] ; col j=(m*2+d), j>=10 -> 0

    const int tid = threadIdx.x;

    // cooperative fill of centers (512 floats, 256 threads)
    lds_c[tid]       = centers[tid];
    lds_c[tid + 256] = centers[tid + 256];

    // cooperative fill of padded weight tile:
    // lds_w[pair*32 + j*2 + parity] = W[m=j>>1][c=2*pair+parity][d=j&1]
    for (int idx = tid; idx < 4096; idx += 256) {
        const int j      = (idx >> 1) & 15;
        const int parity = idx & 1;
        const int c      = ((idx >> 5) << 1) + parity;
        float v = 0.0f;
        if (j < 10) v = weights[(j >> 1) * 512 + c * 2 + (j & 1)];
        lds_w[idx] = v;
    }
    __syncthreads();

    const int lane     = tid & 31;
    const int waveId   = tid >> 5;
    const int row      = lane & 15;   // point row within tile / output column
    const int grp      = lane >> 4;   // K-pair group (0: K={0,1}, 1: K={2,3})
    const int tileBase = (blockIdx.x * 8 + waveId) * 16;

    // load this lane's point (clamped so EXEC stays all-1s for WMMA)
    int p = tileBase + row;
    if (p >= N) p = N - 1;
    const float2 xp = ((const float2*)x)[p];

    v8f acc = {};
    #pragma unroll 4
    for (int k = 0; k < 64; ++k) {
        const int c0 = 4 * k + 2 * grp;               // even -> 16B aligned in lds_c
        const float4 cc = *(const float4*)(&lds_c[c0 * 2]);              // centers c0, c0+1 (broadcast per half-wave)
        const float2 ww = *(const float2*)(&lds_w[c0 * 16 + row * 2]);   // W[c0][j], W[c0+1][j]

        const float dx0 = xp.x - cc.x, dy0 = xp.y - cc.y;
        const float dx1 = xp.x - cc.z, dy1 = xp.y - cc.w;
        const float r0 = SIGMA2 * __expf(-(dx0 * dx0 + dy0 * dy0));
        const float r1 = SIGMA2 * __expf(-(dx1 * dx1 + dy1 * dy1));

        v2f a; a.x = r0;   a.y = r1;     // A: 16x4 f32, K-pair per lane group
        v2f b; b.x = ww.x; b.y = ww.y;   // B: 4x16 f32, same K-pair, col j=row

        // D = A x B + C, full fp32 (8 args: neg_a, A, neg_b, B, c_mod, C, reuse_a, reuse_b)
        acc = __builtin_amdgcn_wmma_f32_16x16x4_f32(
            false, a, false, b, (short)0, acc, false, false);
    }

    // C/D layout: lane L, VGPR v  ->  row M = v + 8*(L>>4), col N = L&15
    const int j = row;               // output column = m*2 + d
    if (j < 10) {
        const int m = j >> 1, d = j & 1;
        float* obase = out + (size_t)m * (size_t)N * 2 + d;
        const int base = tileBase + 8 * grp;
        if (tileBase + 16 <= N) {
            // full tile (wave-uniform branch): unconditional scatter, no per-store EXEC churn
            #pragma unroll
            for (int v = 0; v < 8; ++v)
                obase[(size_t)(base + v) * 2] = acc[v];
        } else {
            // single ragged tail tile
            #pragma unroll
            for (int v = 0; v < 8; ++v)
                if (base + v < N) obase[(size_t)(base + v) * 2] = acc[v];
        }
    }
}

extern "C" void kernel_launch(void* const* d_in, const int* in_sizes, int n_in,
                              void* d_out, int out_size, void* d_ws, size_t ws_size,
                              hipStream_t stream) {
    const float* x       = (const float*)d_in[0];  // [N,2]
    const float* centers = (const float*)d_in[1];  // [256,2]
    const float* weights = (const float*)d_in[2];  // [5,256,2]
    float* out = (float*)d_out;                    // [5,N,2]

    const int N = in_sizes[0] / 2;
    const int pointsPerBlock = 128;                // 8 waves x 16-point WMMA tiles
    const int grid = (N + pointsPerBlock - 1) / pointsPerBlock;

    EnsembleRBF_wmma_kernel<<<grid, 256, 0, stream>>>(x, centers, weights, out, N);
}